// Gaussian_Weighting_86423331930637
// MI455X (gfx1250) — compile-verified
//
#include <hip/hip_runtime.h>
#include <hip/hip_bf16.h>

typedef __attribute__((ext_vector_type(2))) float v2f;
typedef __attribute__((ext_vector_type(8))) float v8f;

// ---- monotonic float <-> uint order encoding (for atomicMax on floats) ----
__device__ __forceinline__ unsigned ordf(float f) {
    unsigned u = __float_as_uint(f);
    return (u >> 31) ? ~u : (u | 0x80000000u);
}
__device__ __forceinline__ float unordf(unsigned u) {
    return (u >> 31) ? __uint_as_float(u & 0x7fffffffu) : __uint_as_float(~u);
}

// ---- 0) zero output + segment accumulators ----
__global__ void k_init(float* __restrict__ out, size_t nout,
                       unsigned* __restrict__ mrows, float* __restrict__ srows, int N) {
    size_t i = (size_t)blockIdx.x * blockDim.x + threadIdx.x;
    size_t stride = (size_t)gridDim.x * blockDim.x;
    for (size_t j = i; j < nout; j += stride) out[j] = 0.0f;
    for (size_t j = i; j < (size_t)N; j += stride) { mrows[j] = 0u; srows[j] = 0.0f; }
}

// ---- 1) per-edge Gaussian log-weight + segment max ----
__global__ void k_weights(const float4* __restrict__ uval, const int* __restrict__ rows,
                          const float* __restrict__ mu, const float* __restrict__ sigma,
                          float* __restrict__ w, unsigned* __restrict__ mrows, int E) {
    int e = blockIdx.x * blockDim.x + threadIdx.x;
    if (e >= E) return;
    float4 u = uval[e];
    float4 m = *(const float4*)mu;     // D == 4
    float4 s = *(const float4*)sigma;
    float i0 = 1.0f / (s.x * s.x + 1e-14f);
    float i1 = 1.0f / (s.y * s.y + 1e-14f);
    float i2 = 1.0f / (s.z * s.z + 1e-14f);
    float i3 = 1.0f / (s.w * s.w + 1e-14f);
    float d0 = u.x - m.x, d1 = u.y - m.y, d2 = u.z - m.z, d3 = u.w - m.w;
    float ww = -0.5f * (d0 * d0 * i0 + d1 * d1 * i1 + d2 * d2 * i2 + d3 * d3 * i3);
    w[e] = ww;
    atomicMax(&mrows[rows[e]], ordf(ww));
}

// ---- 2) exp(w - max) + segment sum ----
__global__ void k_exp(const int* __restrict__ rows, float* __restrict__ w,
                      const unsigned* __restrict__ mrows, float* __restrict__ srows, int E) {
    int e = blockIdx.x * blockDim.x + threadIdx.x;
    if (e >= E) return;
    int r = rows[e];
    float ev = __expf(w[e] - unordf(mrows[r]));
    w[e] = ev;                       // reuse buffer: now holds exp value
    atomicAdd(&srows[r], ev);
}

// ---- 3) WMMA-tiled scatter SpMM: out[b,row_e,:] += (e_v/s[row]) * X[b,col_e,:] ----
// One wave (32 threads) per 16-edge tile per batch. A is a 16x16 selection*softmax
// matrix (dedup within tile), B is 16 gathered X rows from LDS, K=16 via 4x K=4 WMMA.
// A fragments and scatter targets are f0-invariant and hoisted; scatter targets are
// 32-bit element offsets from the uniform `out` base so atomics use SADDR+voffset.
__global__ __launch_bounds__(32) void k_spmm_wmma(
    const float* __restrict__ X, const int* __restrict__ rows, const int* __restrict__ cols,
    const float* __restrict__ expv, const float* __restrict__ srows,
    float* __restrict__ out, int N, int E) {

    __shared__ float sX[16 * 128];   // 16 gathered rows of F=128 floats (8 KB)
    __shared__ int scol[16];

    const int lane = threadIdx.x;    // wave32
    const int tile = blockIdx.x;
    const int b    = blockIdx.y;

    int e = tile * 16 + lane;
    int row = -1;                    // invalid marker (guards scatter)
    float smv = 0.0f;
    if (lane < 16) {
        int col = 0;
        if (e < E) {
            row = rows[e];
            col = cols[e];
            smv = expv[e] / srows[row];
        }
        scol[lane] = col;
    }
    __syncthreads();

    // Stage 16 X rows into LDS: one full 512B row per iteration, coalesced float4.
    const float4* X4 = (const float4*)X;
    float4* sX4 = (float4*)sX;
    #pragma unroll 4
    for (int i = 0; i < 16; ++i) {
        sX4[i * 32 + lane] = X4[((size_t)b * N + scol[i]) * 32 + lane];
    }
    __syncthreads();

    // slot(k) = first edge index in tile with the same row (intra-tile dedup)
    int slot = lane & 15;
    #pragma unroll
    for (int j = 0; j < 16; ++j) {
        int rj = __shfl(row, j);
        if (rj == row && j < slot) slot = j;
    }

    const int half = lane >> 4;      // 0: M/K-low half, 1: M/K-high half
    const int nIdx = lane & 15;      // N index (and A's M index) for this lane

    // ---- hoisted: A fragments (independent of F-chunk) ----
    v2f afr[4];
    #pragma unroll
    for (int t = 0; t < 4; ++t) {
        int j0 = 4 * t + 2 * half;   // edges covered by this lane's K slots
        int j1 = j0 + 1;
        int s0 = __shfl(slot, j0), s1 = __shfl(slot, j1);
        float w0 = __shfl(smv, j0), w1 = __shfl(smv, j1);
        afr[t].x = (s0 == nIdx) ? w0 : 0.0f;   // A[m=nIdx][k]
        afr[t].y = (s1 == nIdx) ? w1 : 0.0f;
    }

    // ---- hoisted: scatter destinations as 32-bit element offsets (output is
    // B*N*F = 12.8M elements, so 32-bit indexing is exact) ----
    unsigned offg[8];
    bool     emit[8];
    const unsigned obase = (unsigned)b * (unsigned)N * 128u + (unsigned)nIdx;
    #pragma unroll
    for (int g = 0; g < 8; ++g) {
        int m  = g + half * 8;           // D layout: VGPR g -> M = g (+8 hi half)
        int rm = __shfl(row, m);
        int sl = __shfl(slot, m);
        emit[g] = (sl == m) && (rm >= 0);
        unsigned rs = (rm < 0) ? 0u : (unsigned)rm;  // safe offset even when not emitted
        offg[g] = obase + rs * 128u;
    }

    for (int f0 = 0; f0 < 128; f0 += 16) {
        v8f acc = {};
        #pragma unroll
        for (int t = 0; t < 4; ++t) {
            int j0 = 4 * t + 2 * half;
            v2f bb;
            bb.x = sX[j0 * 128 + f0 + nIdx];        // B[k][n=nIdx]
            bb.y = sX[(j0 + 1) * 128 + f0 + nIdx];
            acc = __builtin_amdgcn_wmma_f32_16x16x4_f32(
                false, afr[t], false, bb, (short)0, acc, false, false);
        }
        #pragma unroll
        for (int g = 0; g < 8; ++g) {
            if (emit[g]) atomicAdd(out + offg[g] + (unsigned)f0, acc[g]);
        }
    }
}

extern "C" void kernel_launch(void* const* d_in, const int* in_sizes, int n_in,
                              void* d_out, int out_size, void* d_ws, size_t ws_size,
                              hipStream_t stream) {
    const float* X     = (const float*)d_in[0];
    const float* uval  = (const float*)d_in[1];
    const int*   rows  = (const int*)d_in[2];
    const int*   cols  = (const int*)d_in[3];
    const float* mu    = (const float*)d_in[4];
    const float* sigma = (const float*)d_in[5];

    const int E = in_sizes[2];
    const int N = in_sizes[0] / (2 * 128);   // B=2, F=128
    float*    w     = (float*)d_ws;          // E floats: w, then exp values
    unsigned* mrows = (unsigned*)(w + E);    // N uints
    float*    srows = (float*)(mrows + N);   // N floats
    float*    out   = (float*)d_out;
    size_t    nout  = (size_t)out_size;

    k_init<<<8192, 256, 0, stream>>>(out, nout, mrows, srows, N);
    int blocksE = (E + 255) / 256;
    k_weights<<<blocksE, 256, 0, stream>>>((const float4*)uval, rows, mu, sigma, w, mrows, E);
    k_exp<<<blocksE, 256, 0, stream>>>(rows, w, mrows, srows, E);
    dim3 grid((E + 15) / 16, 2);
    k_spmm_wmma<<<grid, 32, 0, stream>>>(X, rows, cols, w, srows, out, N, E);
}